// Criterion_ConfusionClassification_24404004176543
// MI455X (gfx1250) — compile-verified
//
#include <hip/hip_runtime.h>

// CDNA5 (gfx1250) wave32 WMMA types
typedef __attribute__((ext_vector_type(2))) float v2f;
typedef __attribute__((ext_vector_type(8))) float v8f;

#define TOTAL   (64 * 65536)   // B * N
#define THREADS 256            // 8 waves of 32
#define BLOCKS  2048
#define ITERS   (TOTAL / (THREADS * BLOCKS))   // = 8, exact (no tail -> EXEC all-ones)

// Full 32-lane sum via two chained V_WMMA_F32_16X16X4_F32 on the matrix pipe.
// Result is broadcast into every lane (D2[m][n] == total for all m,n).
__device__ __forceinline__ float wave_sum_wmma(float v) {
    // A (16x4 f32, 2 VGPRs): lane L<16 -> A[L][0]=v, A[L][1]=0 ;
    //                        lane L>=16 -> A[L-16][2]=v, A[L-16][3]=0
    v2f a;    a.x = v;    a.y = 0.0f;
    v2f ones; ones.x = 1.0f; ones.y = 1.0f;   // all-ones 16x4 / 4x16 operand
    v8f c = {};
    // D[m][n] = sum_k A[m][k] = v[lane m] + v[lane m+16]   (constant across n)
    v8f d = __builtin_amdgcn_wmma_f32_16x16x4_f32(
        /*neg_a=*/false, a, /*neg_b=*/false, ones,
        /*c_mod=*/(short)0, c, /*reuse_a=*/false, /*reuse_b=*/false);
    // Fold 8 accumulator VGPRs -> 2, arranged as a 4x16 B operand:
    // b2.x: lanes0-15 = s0+s2+s4+s6 (B row0), lanes16-31 = s8+s10+s12+s14 (B row2)
    // b2.y: lanes0-15 = s1+s3+s5+s7 (B row1), lanes16-31 = s9+s11+s13+s15 (B row3)
    v2f b2;
    b2.x = d[0] + d[2] + d[4] + d[6];
    b2.y = d[1] + d[3] + d[5] + d[7];
    // D2[m][n] = B2[0][n]+B2[1][n]+B2[2][n]+B2[3][n] = sum over all 32 lanes of v
    v8f d2 = __builtin_amdgcn_wmma_f32_16x16x4_f32(
        false, ones, false, b2, (short)0, c, false, false);
    return d2[0];
}

// Deterministic block reduction: WMMA wave sums -> LDS -> serial add by thread 0.
__device__ __forceinline__ float block_sum(float v, float* lds) {
    float w = wave_sum_wmma(v);          // every lane holds its wave's total
    const int wave = threadIdx.x >> 5;
    if ((threadIdx.x & 31) == 0) lds[wave] = w;
    __syncthreads();
    float s = 0.0f;
    if (threadIdx.x == 0) {
#pragma unroll
        for (int i = 0; i < THREADS / 32; ++i) s += lds[i];
    }
    return s;   // valid in thread 0 only
}

__global__ void __launch_bounds__(THREADS)
confusion_loss_partial(const float2* __restrict__ logits,   // [B*N] x {c0,c1}
                       const float4* __restrict__ conf4,    // [B*N] x 4 logits
                       const int*    __restrict__ tgt,      // [B*N] in {0,1}
                       float*        __restrict__ partial)  // [BLOCKS]
{
    __shared__ float lds[THREADS / 32];
    const int base   = blockIdx.x * THREADS + threadIdx.x;
    const int stride = BLOCKS * THREADS;

    float acc = 0.0f;
#pragma unroll
    for (int it = 0; it < ITERS; ++it) {
        const int i = base + it * stride;          // uniform trip count, coalesced
        const float2 lg = logits[i];               // global_load_b64
        const float4 x  = conf4[i];                // global_load_b128
        const int    g  = tgt[i];                  // global_load_b32

        // argmax over 2 logits; strict '>' matches jnp.argmax first-max tiebreak
        const int pred = (lg.y > lg.x) ? 1 : 0;
        // TP=1, FN=2, FP=3, TN=0
        const int cidx = (g != 0) ? (pred ? 1 : 2) : (pred ? 3 : 0);

        // stable log-softmax over 4 classes
        const float xm  = fmaxf(fmaxf(x.x, x.y), fmaxf(x.z, x.w));
        const float se  = __expf(x.x - xm) + __expf(x.y - xm) +
                          __expf(x.z - xm) + __expf(x.w - xm);
        const float lse = xm + __logf(se);
        const float xc  = (cidx == 0) ? x.x : (cidx == 1) ? x.y
                        : (cidx == 2) ? x.z : x.w;
        acc += (lse - xc);                         // nll = -(x_c - lse)
    }

    const float bs = block_sum(acc, lds);
    if (threadIdx.x == 0) partial[blockIdx.x] = bs;
}

__global__ void __launch_bounds__(THREADS)
confusion_loss_final(const float* __restrict__ partial,     // [BLOCKS]
                     float*       __restrict__ out)         // [1]
{
    __shared__ float lds[THREADS / 32];
    float acc = 0.0f;
#pragma unroll
    for (int it = 0; it < BLOCKS / THREADS; ++it)            // 8 each, uniform
        acc += partial[it * THREADS + threadIdx.x];

    const float bs = block_sum(acc, lds);
    if (threadIdx.x == 0) out[0] = bs * (1.0f / (float)TOTAL);
}

extern "C" void kernel_launch(void* const* d_in, const int* in_sizes, int n_in,
                              void* d_out, int out_size, void* d_ws, size_t ws_size,
                              hipStream_t stream) {
    (void)in_sizes; (void)n_in; (void)out_size; (void)ws_size;
    const float2* logits = (const float2*)d_in[0];   // pred_logits   [B,N,2] f32
    const float4* confu  = (const float4*)d_in[1];   // pred_confusion[B,N,4] f32
    const int*    tgt    = (const int*)   d_in[2];   // target_classes[B,N]   i32
    float* partial = (float*)d_ws;                   // 2048 floats of scratch

    confusion_loss_partial<<<BLOCKS, THREADS, 0, stream>>>(logits, confu, tgt, partial);
    confusion_loss_final<<<1, THREADS, 0, stream>>>(partial, (float*)d_out);
}